// Nnue_12936441496170
// MI455X (gfx1250) — compile-verified
//
#include <hip/hip_runtime.h>
#include <math.h>

typedef __attribute__((ext_vector_type(2))) float v2f;
typedef __attribute__((ext_vector_type(8))) float v8f;

#define B_    16384
#define BAG_  32
#define D_    256
#define E_    32
#define CLIP_HI_ 0.9921875f   // 127/128
#define LEAK_    0.1f

__device__ __forceinline__ float crelu(float x) {
    float c = fminf(fmaxf(x, 0.0f), CLIP_HI_);
    return c + LEAK_ * (x - c);
}

// ---------------------------------------------------------------------------
// Pass 1: embedding-bag sum + bias, psqt extraction, clipped relu, and
// bucketing of examples by expert. One bag per 64-thread block; each thread
// owns 4 consecutive columns -> fully coalesced GLOBAL_LOAD_B128 from the
// (L2-resident, 109MB) table. Prefetch next row to hide L2 latency.
// ---------------------------------------------------------------------------
__global__ __launch_bounds__(64) void embed_kernel(
    const int* __restrict__ indices, const int* __restrict__ offsets,
    const int* __restrict__ which_model, const int* __restrict__ lengths,
    const float* __restrict__ table, const float* __restrict__ bias,
    float* __restrict__ emb, float* __restrict__ psqt,
    int* __restrict__ cnt, int* __restrict__ bucket)
{
    const int b   = blockIdx.x;
    const int t   = threadIdx.x;
    const int col = t * 4;

    const long off = (long)offsets[b];
    const int  len = lengths[b];
    const int* idxp = indices + off;

    float4 acc = make_float4(0.f, 0.f, 0.f, 0.f);
    for (int j = 0; j < len; ++j) {
        const long idx = (long)idxp[j];
        if (j + 1 < len) {
            // global_prefetch_b8 of next gathered row segment
            __builtin_prefetch(table + (long)idxp[j + 1] * D_ + col, 0, 0);
        }
        float4 v = *(const float4*)(table + idx * D_ + col);
        acc.x += v.x; acc.y += v.y; acc.z += v.z; acc.w += v.w;
    }

    float4 bs = *(const float4*)(bias + col);
    acc.x += bs.x; acc.y += bs.y; acc.z += bs.z; acc.w += bs.w;

    if (t == 0) {
        psqt[b] = acc.x;                       // pre-activation accum[:,0]
        const int e   = which_model[b];
        const int pos = atomicAdd(&cnt[e], 1); // bucket examples per expert
        bucket[e * B_ + pos] = b;
    }

    float4 o;
    o.x = crelu(acc.x); o.y = crelu(acc.y); o.z = crelu(acc.z); o.w = crelu(acc.w);
    *(float4*)(emb + (long)b * D_ + col) = o;
}

// ---------------------------------------------------------------------------
// Pass 2: per-expert MoE MLP using V_WMMA_F32_16X16X4_F32.
// Grid = E * (B/16) single-wave blocks; block (e, tile) handles 16 examples
// of expert e. Layer1: 64 WMMAs over K=256. Layer2: h1 staged through LDS
// into A-layout, 2x4 WMMAs. Layer3 + tanh on lanes 0..15.
// ---------------------------------------------------------------------------
__global__ __launch_bounds__(32) void mlp_kernel(
    const float* __restrict__ W1, const float* __restrict__ b1,
    const float* __restrict__ W2, const float* __restrict__ b2,
    const float* __restrict__ W3, const float* __restrict__ b3,
    const float* __restrict__ emb, const float* __restrict__ psqt,
    const int* __restrict__ cnt, const int* __restrict__ bucket,
    float* __restrict__ out)
{
    __shared__ float h1s[16 * 16];
    __shared__ float h2s[16 * 32];

    const int e      = blockIdx.x >> 10;     // / (B/16)
    const int tile   = blockIdx.x & 1023;
    const int n_e    = cnt[e];
    const int tstart = tile * 16;
    if (tstart >= n_e) return;               // wave-uniform early exit

    const int lane = threadIdx.x;
    const int row  = lane & 15;              // M (A/C) or N (B/C) index
    const int hi   = lane >> 4;              // K-pair selector

    int ridx = tstart + row;
    if (ridx >= n_e) ridx = tstart;          // pad partial tile with row 0
    const int ex = bucket[e * B_ + ridx];

    // ---- Layer 1: (16 x 256) x (256 x 16) ----
    const float* Arow = emb + (long)ex * D_ + 2 * hi;
    const float* Brow = W1 + ((long)e * 16 + row) * D_ + 2 * hi;
    v8f c = {};
    #pragma unroll 4
    for (int k0 = 0; k0 < D_; k0 += 4) {
        v2f a  = *(const v2f*)(Arow + k0);
        v2f bm = *(const v2f*)(Brow + k0);
        c = __builtin_amdgcn_wmma_f32_16x16x4_f32(false, a, false, bm,
                                                  (short)0, c, false, false);
    }
    const float bb1 = b1[e * 16 + row];
    #pragma unroll
    for (int i = 0; i < 8; ++i) {
        const int m = i + 8 * hi;            // C layout: reg i, lane -> (m, row)
        h1s[m * 16 + row] = crelu(c[i] + bb1);
    }
    __syncthreads();

    // ---- Layer 2: (16 x 16) x (16 x 32) as two N=16 halves ----
    v8f c20 = {};
    v8f c21 = {};
    #pragma unroll
    for (int k0 = 0; k0 < 16; k0 += 4) {
        v2f a   = *(const v2f*)(&h1s[row * 16 + k0 + 2 * hi]);
        v2f bm0 = *(const v2f*)(W2 + ((long)e * 32 + row) * 16 + k0 + 2 * hi);
        v2f bm1 = *(const v2f*)(W2 + ((long)e * 32 + 16 + row) * 16 + k0 + 2 * hi);
        c20 = __builtin_amdgcn_wmma_f32_16x16x4_f32(false, a, false, bm0,
                                                    (short)0, c20, false, false);
        c21 = __builtin_amdgcn_wmma_f32_16x16x4_f32(false, a, false, bm1,
                                                    (short)0, c21, false, false);
    }
    const float bb20 = b2[e * 32 + row];
    const float bb21 = b2[e * 32 + 16 + row];
    #pragma unroll
    for (int i = 0; i < 8; ++i) {
        const int m = i + 8 * hi;
        h2s[m * 32 + row]      = crelu(c20[i] + bb20);
        h2s[m * 32 + 16 + row] = crelu(c21[i] + bb21);
    }
    __syncthreads();

    // ---- Layer 3 (dot of 32) + psqt + tanh, lanes 0..15 ----
    if (lane < 16 && (tstart + lane) < n_e) {
        const int m   = lane;
        const int ex2 = bucket[e * B_ + tstart + m];
        float v = b3[e];
        #pragma unroll
        for (int g = 0; g < 32; ++g)
            v += W3[e * 32 + g] * h2s[m * 32 + g];
        out[ex2] = tanhf(v + psqt[ex2]);
    }
}

extern "C" void kernel_launch(void* const* d_in, const int* in_sizes, int n_in,
                              void* d_out, int out_size, void* d_ws, size_t ws_size,
                              hipStream_t stream) {
    const int*   indices     = (const int*)d_in[0];
    const int*   offsets     = (const int*)d_in[1];
    const int*   which_model = (const int*)d_in[2];
    const int*   lengths     = (const int*)d_in[3];
    const float* table       = (const float*)d_in[4];
    const float* bias        = (const float*)d_in[5];
    const float* W1          = (const float*)d_in[6];
    const float* b1          = (const float*)d_in[7];
    const float* W2          = (const float*)d_in[8];
    const float* b2          = (const float*)d_in[9];
    const float* W3          = (const float*)d_in[10];
    const float* b3          = (const float*)d_in[11];
    float* out = (float*)d_out;

    char* ws = (char*)d_ws;
    float* emb    = (float*)ws;                                   // B*D floats
    float* psqt   = (float*)(ws + (size_t)B_ * D_ * 4);           // B floats
    int*   cnt    = (int*)  (ws + (size_t)B_ * D_ * 4 + (size_t)B_ * 4);
    int*   bucket = (int*)  (ws + (size_t)B_ * D_ * 4 + (size_t)B_ * 4 + (size_t)E_ * 4);

    hipMemsetAsync(cnt, 0, E_ * sizeof(int), stream);
    embed_kernel<<<B_, 64, 0, stream>>>(indices, offsets, which_model, lengths,
                                        table, bias, emb, psqt, cnt, bucket);
    mlp_kernel<<<E_ * (B_ / 16), 32, 0, stream>>>(W1, b1, W2, b2, W3, b3,
                                                  emb, psqt, cnt, bucket, out);
}